// CPUReverseForgetMult_1580547973140
// MI455X (gfx1250) — compile-verified
//
#include <hip/hip_runtime.h>
#include <cstdint>
#include <cstddef>

// Problem constants (from the reference): T=2048, B=16, D=1024.
#define T_DIM   2048
#define N_COL   (16 * 1024)          // B*D independent recurrence columns
#define CHUNKS  16                   // time chunks -> 8192 waves of parallelism
#define L_CHUNK (T_DIM / CHUNKS)     // 128 timesteps per chunk
#define BLOCK   256                  // 8 waves per workgroup (wave32)
#define STAGE   8                    // timesteps per LDS stage buffer
#define NSTAGE  (L_CHUNK / STAGE)    // 16 stages per chunk

#if defined(__HIP_DEVICE_COMPILE__) && defined(__gfx1250__) &&          \
    __has_builtin(__builtin_amdgcn_global_load_async_to_lds_b32) &&     \
    __has_builtin(__builtin_amdgcn_s_wait_asynccnt)
#define USE_ASYNC 1
#else
#define USE_ASYNC 0
#endif

#if USE_ASYNC
typedef __attribute__((address_space(1))) int global_int_t;   // builtin arg0 type
typedef __attribute__((address_space(3))) int lds_int_t;      // builtin arg1 type
#endif

// Phase 1 (WRITE_OUT=false): per-chunk reduction  A = prod(1-f), H = local scan.
// Phase 3 (WRITE_OUT=true):  rerun recurrence seeded with carry, write output.
template <bool WRITE_OUT>
__global__ __launch_bounds__(BLOCK) void scan_chunk(
    const float* __restrict__ f, const float* __restrict__ x,
    float* __restrict__ out, const float* __restrict__ carry,
    float* __restrict__ Aout, float* __restrict__ Hout)
{
  const int tid = threadIdx.x;
  const int c   = blockIdx.x * BLOCK + tid;   // column (contiguous -> coalesced)
  const int ci  = blockIdx.y;                 // time chunk
  const int t0  = ci * L_CHUNK;

  float h = 0.0f;
  if (WRITE_OUT) h = carry[(size_t)ci * N_COL + c];
  float a = 1.0f;

#if USE_ASYNC
  __shared__ float sf[2][STAGE * BLOCK];
  __shared__ float sx[2][STAGE * BLOCK];

  auto issue = [&](int k, int buf) {
    const size_t base = (size_t)(t0 + k * STAGE) * N_COL + c;
#pragma unroll
    for (int j = 0; j < STAGE; ++j) {
      __builtin_amdgcn_global_load_async_to_lds_b32(
          (global_int_t*)(f + base + (size_t)j * N_COL),
          (lds_int_t*)&sf[buf][j * BLOCK + tid],
          0, 0);
      __builtin_amdgcn_global_load_async_to_lds_b32(
          (global_int_t*)(x + base + (size_t)j * N_COL),
          (lds_int_t*)&sx[buf][j * BLOCK + tid],
          0, 0);
    }
  };

  int buf = 0;
  issue(NSTAGE - 1, 0);                       // prime the pipeline
  for (int k = NSTAGE - 1; k >= 0; --k) {
    if (k > 0) {
      issue(k - 1, buf ^ 1);                  // prefetch next stage first
      // Async loads complete in order: <=16 outstanding means the 16 loads of
      // the current buffer are done while the next 16 remain in flight.
      __builtin_amdgcn_s_wait_asynccnt(2 * STAGE);
    } else {
      __builtin_amdgcn_s_wait_asynccnt(0);
    }
#pragma unroll
    for (int j = STAGE - 1; j >= 0; --j) {    // reverse time within stage
      const float ft = sf[buf][j * BLOCK + tid];
      const float xt = sx[buf][j * BLOCK + tid];
      h = ft * xt + (1.0f - ft) * h;
      if (!WRITE_OUT) a *= (1.0f - ft);
      if (WRITE_OUT) {
        __builtin_nontemporal_store(
            h, &out[(size_t)(t0 + k * STAGE + j) * N_COL + c]);
      }
    }
    buf ^= 1;
  }
#else
  // Fallback: register double-buffering (still unrolled, 16 loads in flight).
  for (int k = NSTAGE - 1; k >= 0; --k) {
    const size_t base = (size_t)(t0 + k * STAGE) * N_COL + c;
    float rf[STAGE], rx[STAGE];
#pragma unroll
    for (int j = 0; j < STAGE; ++j) {
      rf[j] = f[base + (size_t)j * N_COL];
      rx[j] = x[base + (size_t)j * N_COL];
    }
#pragma unroll
    for (int j = STAGE - 1; j >= 0; --j) {
      h = rf[j] * rx[j] + (1.0f - rf[j]) * h;
      if (!WRITE_OUT) a *= (1.0f - rf[j]);
      if (WRITE_OUT)
        __builtin_nontemporal_store(h, &out[base + (size_t)j * N_COL]);
    }
  }
#endif

  if (!WRITE_OUT) {
    Aout[(size_t)ci * N_COL + c] = a;
    Hout[(size_t)ci * N_COL + c] = h;
  }
}

// Phase 2: sequential combine across chunks (reverse time).
// carry[ci] = H[ci+1] + A[ci+1] * carry[ci+1], carry[CHUNKS-1] = 0.
__global__ __launch_bounds__(BLOCK) void combine_chunks(
    const float* __restrict__ A, const float* __restrict__ H,
    float* __restrict__ carry)
{
  const int c = blockIdx.x * BLOCK + threadIdx.x;
  float cv = 0.0f;
  carry[(size_t)(CHUNKS - 1) * N_COL + c] = cv;
  for (int ci = CHUNKS - 2; ci >= 0; --ci) {
    const size_t idx = (size_t)(ci + 1) * N_COL + c;
    cv = H[idx] + A[idx] * cv;
    carry[(size_t)ci * N_COL + c] = cv;
  }
}

// Safety fallback if the harness workspace is too small: one thread per column
// over all T (latency-limited but correct, and needs no scratch).
__global__ __launch_bounds__(BLOCK) void naive_scan(
    const float* __restrict__ f, const float* __restrict__ x,
    float* __restrict__ out)
{
  const int c = blockIdx.x * BLOCK + threadIdx.x;
  float h = 0.0f;
  for (int t = T_DIM - 1; t >= 0; --t) {
    const size_t idx = (size_t)t * N_COL + c;
    const float ft = f[idx];
    h = ft * x[idx] + (1.0f - ft) * h;
    out[idx] = h;
  }
}

extern "C" void kernel_launch(void* const* d_in, const int* in_sizes, int n_in,
                              void* d_out, int out_size, void* d_ws, size_t ws_size,
                              hipStream_t stream)
{
  const float* f = (const float*)d_in[0];
  const float* x = (const float*)d_in[1];
  float* out = (float*)d_out;

  const size_t need = (size_t)3 * CHUNKS * N_COL * sizeof(float);  // 3 MiB
  if (d_ws == nullptr || ws_size < need) {
    naive_scan<<<dim3(N_COL / BLOCK), dim3(BLOCK), 0, stream>>>(f, x, out);
    return;
  }

  float* A     = (float*)d_ws;
  float* H     = A + (size_t)CHUNKS * N_COL;
  float* carry = H + (size_t)CHUNKS * N_COL;

  dim3 grid(N_COL / BLOCK, CHUNKS);   // 64 x 16 = 1024 WGs = 8192 waves
  scan_chunk<false><<<grid, dim3(BLOCK), 0, stream>>>(f, x, nullptr, nullptr, A, H);
  combine_chunks<<<dim3(N_COL / BLOCK), dim3(BLOCK), 0, stream>>>(A, H, carry);
  scan_chunk<true><<<grid, dim3(BLOCK), 0, stream>>>(f, x, out, carry, nullptr, nullptr);
}